// JointCrossAttention_67585605369938
// MI455X (gfx1250) — compile-verified
//
#include <hip/hip_runtime.h>
#include <math.h>

// Problem constants (match reference).
#define E_DIM   256
#define F_DIM   768
#define BATCH   256
#define MROWS   32    // rows per fused block (two 16-row WMMA stripes)
#define LDSS    260   // padded LDS row stride in floats: banks = (4*m + k) % 64, conflict-free

typedef __attribute__((ext_vector_type(2))) float v2f;
typedef __attribute__((ext_vector_type(8))) float v8f;

// ---------------------------------------------------------------------------
// V_WMMA_F32_16X16X4_F32 lane mapping (ISA 7.12.2, wave32):
//   A (16x4, MxK):  lane holds row M = lane&15; VGPR pair = K = 2*(lane>>4)+{0,1}
//   B (4x16, KxN):  lane holds col N = lane&15; VGPR pair = K = 2*(lane>>4)+{0,1}
//   C/D (16x16):    acc[v] = element (M = v + 8*(lane>>4), N = lane&15)
// ---------------------------------------------------------------------------

// Dual-stripe K=256 loop: A is a [32 x 256] LDS stripe (rows 0..15 -> acc0,
// rows 16..31 -> acc1); each B fragment (global, row of W) feeds TWO WMMAs.
__device__ __forceinline__ void kloop256_dual(const float (*src)[LDSS],
                                              const float* __restrict__ wrow,
                                              int m, int kh,
                                              v8f& acc0, v8f& acc1) {
#pragma unroll 8
  for (int kb = 0; kb < 256; kb += 4) {
    v2f b  = *(const v2f*)(wrow + kb + kh);
    v2f a0 = *(const v2f*)(&src[m][kb + kh]);
    v2f a1 = *(const v2f*)(&src[m + 16][kb + kh]);
    acc0 = __builtin_amdgcn_wmma_f32_16x16x4_f32(false, a0, false, b,
                                                 (short)0, acc0, false, false);
    acc1 = __builtin_amdgcn_wmma_f32_16x16x4_f32(false, a1, false, b,
                                                 (short)0, acc1, false, false);
  }
}

// ---------------------------------------------------------------------------
// Phase 1: generic C[256,256] = X[256,K] @ W[256,K]^T (+ bias), fp32 WMMA.
// 256 output tiles of 16x16; 4 waves/block -> 64 blocks.
// ---------------------------------------------------------------------------
__global__ __launch_bounds__(128) void gemm_xwT(const float* __restrict__ X,
                                                const float* __restrict__ W,
                                                const float* __restrict__ bias,
                                                float* __restrict__ C, int K) {
  const int lane = threadIdx.x & 31;
  const int wave = threadIdx.x >> 5;
  const int tile = blockIdx.x * 4 + wave;       // 0..255
  const int m0   = (tile >> 4) << 4;
  const int n0   = (tile & 15) << 4;
  const int m    = lane & 15;
  const int kh   = (lane >> 4) << 1;            // 0 or 2
  const int mhi  = (lane >> 4) << 3;            // 0 or 8
  const int col  = n0 + m;

  v8f acc;
  const float bv = bias ? bias[col] : 0.0f;
#pragma unroll
  for (int v = 0; v < 8; ++v) acc[v] = bv;

  const float* __restrict__ arow = X + (size_t)(m0 + m) * K;
  const float* __restrict__ wrow = W + (size_t)col * K;
#pragma unroll 8
  for (int kb = 0; kb < K; kb += 4) {
    v2f a = *(const v2f*)(arow + kb + kh);
    v2f b = *(const v2f*)(wrow + kb + kh);
    acc = __builtin_amdgcn_wmma_f32_16x16x4_f32(false, a, false, b,
                                                (short)0, acc, false, false);
  }

#pragma unroll
  for (int v = 0; v < 8; ++v)
    C[(size_t)(m0 + mhi + v) * E_DIM + col] = acc[v];
}

// ---------------------------------------------------------------------------
// Phase 2: fused per-(batch, 32-row-stripe) chain. Grid = BATCH * (E/32) =
// 2048 blocks of 128 threads (4 waves; wave w owns N-tiles w*4..w*4+3, and
// computes BOTH 16-row stripes per tile so every weight fragment feeds 2 WMMAs).
//
//   buf0 <- tanh(enc1[b,i] * aff_a[b,j] / 16)                  (elementwise)
//   buf1 <- relu(buf0 @ wca^T + Wa[i,j])                        (WMMA)
//   bufA1<- buf1 @ wha^T + enc1[i,j]                            (WMMA)
//   buf0 <- tanh(enc2[b,i] * aff_v[b,j] / 16)
//   buf1 <- relu(buf0 @ wcv^T + Wv[i,j])
//   buf0 <- buf1 @ whv^T + enc2[i,j]                            (= A2)
//   buf1 <- relu(bufA1 @ fc1w[:, :256]^T + buf0 @ fc1w[:,256:]^T + fc1_b[j])
//   out[b, i0+r] = sum_j buf1[r][j] * fc2w[j] + fc2b
// ---------------------------------------------------------------------------
__global__ __launch_bounds__(128) void fused_chain(
    const float* __restrict__ enc1, const float* __restrict__ enc2,
    const float* __restrict__ affa, const float* __restrict__ affv,
    const float* __restrict__ Wa,   const float* __restrict__ Wv,
    const float* __restrict__ wca,  const float* __restrict__ wcv,
    const float* __restrict__ wha,  const float* __restrict__ whv,
    const float* __restrict__ fc1w, const float* __restrict__ fc1b,
    const float* __restrict__ fc2w, const float* __restrict__ fc2b,
    float* __restrict__ out) {
  __shared__ float buf0[MROWS][LDSS];
  __shared__ float buf1[MROWS][LDSS];
  __shared__ float bufA1[MROWS][LDSS];
  __shared__ float red[MROWS][4];

  const int tid  = threadIdx.x;
  const int lane = tid & 31;
  const int wave = tid >> 5;
  const int b    = blockIdx.x >> 3;          // batch
  const int i0   = (blockIdx.x & 7) << 5;    // row stripe start (32 rows)
  const int m    = lane & 15;
  const int kh   = (lane >> 4) << 1;
  const int mhi  = (lane >> 4) << 3;
  const float scale = 0.0625f;               // 1/sqrt(256)

  // ---- stage 1: attn_a stripe (elementwise tanh of outer product) ----
  for (int idx = tid; idx < MROWS * E_DIM; idx += 128) {
    const int r = idx >> 8, j = idx & 255;
    buf0[r][j] = tanhf(enc1[b * E_DIM + i0 + r] * affa[b * E_DIM + j] * scale);
  }
  __syncthreads();

  // ---- stage 2: H_a = relu(attn_a @ wca^T + Wa stripe) ----
#pragma unroll
  for (int t = 0; t < 4; ++t) {
    const int col = (wave * 4 + t) * 16 + m;
    v8f acc0, acc1;
#pragma unroll
    for (int v = 0; v < 8; ++v) {
      acc0[v] = Wa[(size_t)(i0 + mhi + v) * E_DIM + col];
      acc1[v] = Wa[(size_t)(i0 + 16 + mhi + v) * E_DIM + col];
    }
    kloop256_dual(buf0, wca + (size_t)col * E_DIM, m, kh, acc0, acc1);
#pragma unroll
    for (int v = 0; v < 8; ++v) {
      buf1[mhi + v][col]      = fmaxf(acc0[v], 0.0f);
      buf1[16 + mhi + v][col] = fmaxf(acc1[v], 0.0f);
    }
  }
  __syncthreads();

  // ---- stage 3: A1 = H_a @ wha^T + enc1[i,j] ----
#pragma unroll
  for (int t = 0; t < 4; ++t) {
    const int col = (wave * 4 + t) * 16 + m;
    v8f acc0, acc1;
#pragma unroll
    for (int v = 0; v < 8; ++v) {
      acc0[v] = enc1[(size_t)(i0 + mhi + v) * E_DIM + col];
      acc1[v] = enc1[(size_t)(i0 + 16 + mhi + v) * E_DIM + col];
    }
    kloop256_dual(buf1, wha + (size_t)col * E_DIM, m, kh, acc0, acc1);
#pragma unroll
    for (int v = 0; v < 8; ++v) {
      bufA1[mhi + v][col]      = acc0[v];
      bufA1[16 + mhi + v][col] = acc1[v];
    }
  }
  __syncthreads();

  // ---- stage 4: attn_v stripe ----
  for (int idx = tid; idx < MROWS * E_DIM; idx += 128) {
    const int r = idx >> 8, j = idx & 255;
    buf0[r][j] = tanhf(enc2[b * E_DIM + i0 + r] * affv[b * E_DIM + j] * scale);
  }
  __syncthreads();

  // ---- stage 5: H_v = relu(attn_v @ wcv^T + Wv stripe) ----
#pragma unroll
  for (int t = 0; t < 4; ++t) {
    const int col = (wave * 4 + t) * 16 + m;
    v8f acc0, acc1;
#pragma unroll
    for (int v = 0; v < 8; ++v) {
      acc0[v] = Wv[(size_t)(i0 + mhi + v) * E_DIM + col];
      acc1[v] = Wv[(size_t)(i0 + 16 + mhi + v) * E_DIM + col];
    }
    kloop256_dual(buf0, wcv + (size_t)col * E_DIM, m, kh, acc0, acc1);
#pragma unroll
    for (int v = 0; v < 8; ++v) {
      buf1[mhi + v][col]      = fmaxf(acc0[v], 0.0f);
      buf1[16 + mhi + v][col] = fmaxf(acc1[v], 0.0f);
    }
  }
  __syncthreads();

  // ---- stage 6: A2 = H_v @ whv^T + enc2[i,j] -> buf0 ----
#pragma unroll
  for (int t = 0; t < 4; ++t) {
    const int col = (wave * 4 + t) * 16 + m;
    v8f acc0, acc1;
#pragma unroll
    for (int v = 0; v < 8; ++v) {
      acc0[v] = enc2[(size_t)(i0 + mhi + v) * E_DIM + col];
      acc1[v] = enc2[(size_t)(i0 + 16 + mhi + v) * E_DIM + col];
    }
    kloop256_dual(buf1, whv + (size_t)col * E_DIM, m, kh, acc0, acc1);
#pragma unroll
    for (int v = 0; v < 8; ++v) {
      buf0[mhi + v][col]      = acc0[v];
      buf0[16 + mhi + v][col] = acc1[v];
    }
  }
  __syncthreads();

  // ---- stage 7: h = relu(A1 @ fc1w[:, :256]^T + A2 @ fc1w[:, 256:]^T + b) ----
#pragma unroll
  for (int t = 0; t < 4; ++t) {
    const int col = (wave * 4 + t) * 16 + m;
    v8f acc0, acc1;
    const float bv = fc1b[col];
#pragma unroll
    for (int v = 0; v < 8; ++v) { acc0[v] = bv; acc1[v] = bv; }
    kloop256_dual(bufA1, fc1w + (size_t)col * (2 * E_DIM),         m, kh, acc0, acc1);
    kloop256_dual(buf0,  fc1w + (size_t)col * (2 * E_DIM) + E_DIM, m, kh, acc0, acc1);
#pragma unroll
    for (int v = 0; v < 8; ++v) {
      buf1[mhi + v][col]      = fmaxf(acc0[v], 0.0f);
      buf1[16 + mhi + v][col] = fmaxf(acc1[v], 0.0f);
    }
  }
  __syncthreads();

  // ---- stage 8: out[b, i0+r] = <h[r,:], fc2w> + fc2b ----
  // 4 threads per row, 64 columns each, combine via LDS.
  {
    const int r = tid >> 2;          // 0..31
    const int q = tid & 3;           // 0..3
    float s = 0.0f;
#pragma unroll 4
    for (int j = q * 64; j < (q + 1) * 64; ++j) s += buf1[r][j] * fc2w[j];
    red[r][q] = s;
  }
  __syncthreads();
  if (tid < MROWS) {
    out[b * E_DIM + i0 + tid] =
        red[tid][0] + red[tid][1] + red[tid][2] + red[tid][3] + fc2b[0];
  }
}

// ---------------------------------------------------------------------------
extern "C" void kernel_launch(void* const* d_in, const int* in_sizes, int n_in,
                              void* d_out, int out_size, void* d_ws, size_t ws_size,
                              hipStream_t stream) {
  const float* features1 = (const float*)d_in[0];
  const float* features2 = (const float*)d_in[1];
  const float* enc1_w    = (const float*)d_in[2];
  const float* enc1_b    = (const float*)d_in[3];
  const float* enc2_w    = (const float*)d_in[4];
  const float* enc2_b    = (const float*)d_in[5];
  const float* affa_w    = (const float*)d_in[6];
  const float* affv_w    = (const float*)d_in[7];
  const float* wa_w      = (const float*)d_in[8];
  const float* wv_w      = (const float*)d_in[9];
  const float* wca_w     = (const float*)d_in[10];
  const float* wcv_w     = (const float*)d_in[11];
  const float* wha_w     = (const float*)d_in[12];
  const float* whv_w     = (const float*)d_in[13];
  const float* fc1_w     = (const float*)d_in[14];
  const float* fc1_b     = (const float*)d_in[15];
  const float* fc2_w     = (const float*)d_in[16];
  const float* fc2_b     = (const float*)d_in[17];

  float* ws   = (float*)d_ws;
  const size_t MAT = (size_t)E_DIM * E_DIM;  // 65536 floats
  float* enc1 = ws + 0 * MAT;
  float* enc2 = ws + 1 * MAT;
  float* affa = ws + 2 * MAT;
  float* affv = ws + 3 * MAT;
  float* Wa   = ws + 4 * MAT;
  float* Wv   = ws + 5 * MAT;

  const dim3 blk(128);

  // Phase 1: small GEMMs into workspace (all WMMA fp32).
  gemm_xwT<<<64, blk, 0, stream>>>(features1, enc1_w, enc1_b, enc1, F_DIM);
  gemm_xwT<<<64, blk, 0, stream>>>(features2, enc2_w, enc2_b, enc2, F_DIM);
  gemm_xwT<<<64, blk, 0, stream>>>(enc1, affa_w, nullptr, affa, E_DIM);
  gemm_xwT<<<64, blk, 0, stream>>>(enc2, affv_w, nullptr, affv, E_DIM);
  gemm_xwT<<<64, blk, 0, stream>>>(enc1, wa_w,   nullptr, Wa,   E_DIM);
  gemm_xwT<<<64, blk, 0, stream>>>(enc2, wv_w,   nullptr, Wv,   E_DIM);

  // Phase 2: fused per-(batch, 32-row-stripe) chain. 256 batches * 8 stripes.
  fused_chain<<<BATCH * (E_DIM / MROWS), blk, 0, stream>>>(
      enc1, enc2, affa, affv, Wa, Wv, wca_w, wcv_w, wha_w, whv_w,
      fc1_w, fc1_b, fc2_w, fc2_b, (float*)d_out);
}